// GLA_26611617366260
// MI455X (gfx1250) — compile-verified
//
#include <hip/hip_runtime.h>

// ============================================================================
// GLA (global-local contextual attention) forward — MI455X / gfx1250.
// wave32, WMMA f32 16x16x4 for all GEMMs (f32 exactness through two chained
// softmaxes + L2 norm), global_prefetch for K-tile streaming.
//
// LDS tiles are k-pair interleaved so every WMMA A/B fragment is one aligned
// ds_load_b64 into an even VGPR pair (no v_mov fixups in the inner loop).
//
// Workspace layout (floats, ~102 MB required):
//   bg, qp, kp, vp, attout, fg : 64*4096 each
//   ckmat, fmcol, S            : 4096*576 each
//   mm                         : 4096
//   wT (transposed patch W)    : 3*64*64
//   big (scores / cr / att)    : 4096*4096  (reused per sample)
// ============================================================================

typedef __attribute__((ext_vector_type(2))) float v2f;
typedef __attribute__((ext_vector_type(8))) float v8f;

#define WMMA4(A, B, C) \
  __builtin_amdgcn_wmma_f32_16x16x4_f32(false, (A), false, (B), (short)0, (C), false, false)

// ---------------------------------------------------------------------------
// Generic WMMA f32 GEMM:  C[M][N] = op(A)*op(B) (+ bias[m])
//   !transA: A row-major [M][K];  transA: A stored [K][M]
//   !transB: B row-major [K][N];  transB: B stored [N][K]
// Requires M,N multiples of 64 and K multiple of 16 (true for all calls here).
// ---------------------------------------------------------------------------
__global__ __launch_bounds__(128)
void gemm_wmma_f32(const float* __restrict__ A, const float* __restrict__ B,
                   const float* __restrict__ bias, float* __restrict__ C,
                   int M, int N, int K, int transA, int transB)
{
  // [kpair][i*2 + klo]; stride 130 floats: even (keeps 8B alignment for b64
  // loads) and skews banks by 2 per kpair row. Lane i reads banks {2i,2i+1}.
  __shared__ __align__(16) float As[8][130];
  __shared__ __align__(16) float Bs[8][130];
  const int tid  = threadIdx.x;
  const int lane = tid & 31;
  const int wave = tid >> 5;
  const int hi   = lane >> 4;    // half-wave select
  const int ml   = lane & 15;
  const int m0   = blockIdx.y * 64;
  const int n0   = blockIdx.x * 64;
  const int m0w  = wave * 16;    // each wave: 16 rows x 64 cols

  v8f acc0 = {}; v8f acc1 = {}; v8f acc2 = {}; v8f acc3 = {};

  const int KT = K >> 4;
  for (int kt = 0; kt < KT; ++kt) {
    const int k0 = kt << 4;
    // ---- stage A tile -> As[k>>1][m*2 + (k&1)]
    if (!transA) {
      const int m = tid >> 1, kh = (tid & 1) << 3;
      const float* g = A + (size_t)(m0 + m) * K + (k0 + kh);
      const float4 x0 = ((const float4*)g)[0];
      const float4 x1 = ((const float4*)g)[1];
      float* d = &As[kh >> 1][m * 2];      // kh even; 4 kpair rows
      d[0] = x0.x; d[1] = x0.y;
      d[130] = x0.z; d[131] = x0.w;
      d[260] = x1.x; d[261] = x1.y;
      d[390] = x1.z; d[391] = x1.w;
      if (kt + 1 < KT) __builtin_prefetch((const void*)(g + 16), 0, 1);
    } else {
      const int k = tid >> 3, m8 = (tid & 7) << 3;
      const float* g = A + (size_t)(k0 + k) * M + (m0 + m8);
      const float4 x0 = ((const float4*)g)[0];
      const float4 x1 = ((const float4*)g)[1];
      float* d = &As[k >> 1][m8 * 2 + (k & 1)];
      d[ 0] = x0.x; d[ 2] = x0.y; d[ 4] = x0.z; d[ 6] = x0.w;
      d[ 8] = x1.x; d[10] = x1.y; d[12] = x1.z; d[14] = x1.w;
      if (kt + 1 < KT) __builtin_prefetch((const void*)(g + (size_t)16 * M), 0, 1);
    }
    // ---- stage B tile -> Bs[k>>1][n*2 + (k&1)]
    if (!transB) {
      const int k = tid >> 3, n8 = (tid & 7) << 3;
      const float* g = B + (size_t)(k0 + k) * N + (n0 + n8);
      const float4 x0 = ((const float4*)g)[0];
      const float4 x1 = ((const float4*)g)[1];
      float* d = &Bs[k >> 1][n8 * 2 + (k & 1)];
      d[ 0] = x0.x; d[ 2] = x0.y; d[ 4] = x0.z; d[ 6] = x0.w;
      d[ 8] = x1.x; d[10] = x1.y; d[12] = x1.z; d[14] = x1.w;
      if (kt + 1 < KT) __builtin_prefetch((const void*)(g + (size_t)16 * N), 0, 1);
    } else {
      const int n = tid >> 1, kh = (tid & 1) << 3;
      const float* g = B + (size_t)(n0 + n) * K + (k0 + kh);
      const float4 x0 = ((const float4*)g)[0];
      const float4 x1 = ((const float4*)g)[1];
      float* d = &Bs[kh >> 1][n * 2];
      d[0] = x0.x; d[1] = x0.y;
      d[130] = x0.z; d[131] = x0.w;
      d[260] = x1.x; d[261] = x1.y;
      d[390] = x1.z; d[391] = x1.w;
      if (kt + 1 < KT) __builtin_prefetch((const void*)(g + 16), 0, 1);
    }
    __syncthreads();

    // A 16x4 frag: lanes 0-15 K={kk,kk+1}, lanes 16-31 K={kk+2,kk+3}; M=ml
    // B 4x16 frag: mirrored; N=ml. Each frag = one aligned ds_load_b64.
#pragma unroll
    for (int kk = 0; kk < 16; kk += 4) {
      const int kp = (kk >> 1) + hi;     // kpair row holding {kb, kb+1}
      const v2f a  = *(const v2f*)&As[kp][(m0w + ml) * 2];
      const v2f b0 = *(const v2f*)&Bs[kp][( 0 + ml) * 2];
      const v2f b1 = *(const v2f*)&Bs[kp][(16 + ml) * 2];
      const v2f b2 = *(const v2f*)&Bs[kp][(32 + ml) * 2];
      const v2f b3 = *(const v2f*)&Bs[kp][(48 + ml) * 2];
      acc0 = WMMA4(a, b0, acc0);
      acc1 = WMMA4(a, b1, acc1);
      acc2 = WMMA4(a, b2, acc2);
      acc3 = WMMA4(a, b3, acc3);
    }
    __syncthreads();
  }

  // C/D layout: vgpr r -> row r (lanes 0-15) / row 8+r (lanes 16-31), col = ml
  const int mr = m0 + m0w + (hi << 3);
#pragma unroll
  for (int r = 0; r < 8; ++r) {
    const int m = mr + r;
    const float bv = bias ? bias[m] : 0.f;
    float* crow = C + (size_t)m * N + n0;
    crow[ 0 + ml] = acc0[r] + bv;
    crow[16 + ml] = acc1[r] + bv;
    crow[32 + ml] = acc2[r] + bv;
    crow[48 + ml] = acc3[r] + bv;
  }
}

// ---------------------------------------------------------------------------
// Small kernels
// ---------------------------------------------------------------------------
__global__ void make_background(const float* __restrict__ F, const float* __restrict__ Mk,
                                float* __restrict__ bg)
{
  const int idx = blockIdx.x * 256 + threadIdx.x;   // 64*4096
  bg[idx] = F[idx] * (1.f - Mk[idx & 4095]);
}

__global__ void compose_fg(const float* __restrict__ F, const float* __restrict__ Mk,
                           const float* __restrict__ attout, float* __restrict__ fg)
{
  const int idx = blockIdx.x * 256 + threadIdx.x;   // 64*4096
  const float m = Mk[idx & 4095];
  fg[idx] = F[idx] * m + (1.f - m) * attout[idx];
}

// rows of `scores` scaled by mask[i], then softmax over j
__global__ void row_softmax_mask(float* __restrict__ S, const float* __restrict__ mrow, int N)
{
  __shared__ float red[256];
  const int i = blockIdx.x, tid = threadIdx.x;
  const float scale = mrow[i];
  float* row = S + (size_t)i * N;

  float mx = -3.4e38f;
  for (int j = tid; j < N; j += 256) mx = fmaxf(mx, row[j] * scale);
  red[tid] = mx; __syncthreads();
  for (int s = 128; s > 0; s >>= 1) { if (tid < s) red[tid] = fmaxf(red[tid], red[tid+s]); __syncthreads(); }
  mx = red[0]; __syncthreads();

  float sum = 0.f;
  for (int j = tid; j < N; j += 256) sum += __expf(row[j] * scale - mx);
  red[tid] = sum; __syncthreads();
  for (int s = 128; s > 0; s >>= 1) { if (tid < s) red[tid] += red[tid+s]; __syncthreads(); }
  const float inv = 1.f / red[0];

  for (int j = tid; j < N; j += 256) row[j] = __expf(row[j] * scale - mx) * inv;
}

// in-place 3x3 box sum on each 64x64 "image" row of CR (one block per row n)
__global__ void box3x3_inplace(float* __restrict__ CR)
{
  __shared__ float img[4096];
  const size_t base = (size_t)blockIdx.x * 4096;
  for (int t = threadIdx.x; t < 4096; t += 256) img[t] = CR[base + t];
  __syncthreads();
  for (int t = threadIdx.x; t < 4096; t += 256) {
    const int y = t >> 6, x = t & 63;
    float s = 0.f;
#pragma unroll
    for (int dy = -1; dy <= 1; ++dy)
#pragma unroll
      for (int dx = -1; dx <= 1; ++dx) {
        const int yy = y + dy, xx = x + dx;
        if ((unsigned)yy < 64u && (unsigned)xx < 64u) s += img[yy * 64 + xx];
      }
    CR[base + t] = s;
  }
}

// att[n][p] = mm[n] * softmax_n( CR[n][p] * mm[n] ), in place. One column per thread.
__global__ void col_softmax_mm(float* __restrict__ CR, const float* __restrict__ mm)
{
  const int p = blockIdx.x * 256 + threadIdx.x;
  float m = -3.4e38f, l = 0.f;
  for (int n = 0; n < 4096; ++n) {
    const float x = CR[(size_t)n * 4096 + p] * mm[n];
    const float nm = fmaxf(m, x);
    l = l * __expf(m - nm) + __expf(x - nm);
    m = nm;
  }
  const float inv = 1.f / l;
  for (int n = 0; n < 4096; ++n) {
    const size_t idx = (size_t)n * 4096 + p;
    const float x = CR[idx] * mm[n];
    CR[idx] = mm[n] * __expf(x - m) * inv;
  }
}

// im2col (3x3, pad 1): col[p][c*9+u*3+v] = fg[c][y+u-1][x+v-1]
__global__ void im2col3(const float* __restrict__ fg, float* __restrict__ col)
{
  const int idx = blockIdx.x * 256 + threadIdx.x;  // 4096*64
  const int p = idx >> 6, c = idx & 63;
  const int y = p >> 6, x = p & 63;
  float* o = col + (size_t)p * 576 + c * 9;
  const float* f = fg + (size_t)c * 4096;
#pragma unroll
  for (int u = 0; u < 3; ++u)
#pragma unroll
    for (int v = 0; v < 3; ++v) {
      const int yy = y + u - 1, xx = x + v - 1;
      o[u * 3 + v] = ((unsigned)yy < 64u && (unsigned)xx < 64u) ? f[yy * 64 + xx] : 0.f;
    }
}

// wT[m][c2][c] = W_m[c][c2]  for coalesced per-channel reads in gap_patches
__global__ void transpose_w(const float* __restrict__ Wq, const float* __restrict__ Wk,
                            const float* __restrict__ Wv, float* __restrict__ wT)
{
  const int idx = blockIdx.x * 256 + threadIdx.x;  // 3*4096
  const int m = idx >> 12, j = idx & 4095;
  const float* W = (m == 0) ? Wq : ((m == 1) ? Wk : Wv);
  wT[m * 4096 + j] = W[(j & 63) * 64 + (j >> 6)];
}

// GlobalAttentionPatch, collapsed: since mk is the 1-channel mask broadcast,
// (qf kf^T) mf  ==  mp[p] * <q[c,:], ksum>, ksum[p] = sum_d k[d][p].
// One 64-thread block per patch n; thread = channel c. Emits normalized
// ckmat[n][c*9+uv] and mm[n].
__global__ __launch_bounds__(64)
void gap_patches(const float* __restrict__ bg, const float* __restrict__ Mk,
                 const float* __restrict__ WqT, const float* __restrict__ WkT,
                 const float* __restrict__ WvT,
                 const float* __restrict__ bq, const float* __restrict__ bk,
                 const float* __restrict__ bv, const float* __restrict__ gamma,
                 float* __restrict__ ckmat, float* __restrict__ mmv)
{
  __shared__ float ckp[64][9];
  __shared__ float ks[64][9];
  __shared__ float mp[9];
  __shared__ float ksum[9];
  __shared__ float red[64];
  const int n = blockIdx.x, c = threadIdx.x;
  const int py = n >> 6, px = n & 63;

  {
    const float* f = bg + (size_t)c * 4096;
#pragma unroll
    for (int u = 0; u < 3; ++u)
#pragma unroll
      for (int v = 0; v < 3; ++v) {
        const int yy = py + u - 1, xx = px + v - 1;
        const float val = ((unsigned)yy < 64u && (unsigned)xx < 64u) ? f[yy * 64 + xx] : 0.f;
        ckp[c][u * 3 + v] = val + 1e-7f;   // ck = patches + 1e-7 (padding included)
      }
  }
  if (c < 9) {
    const int u = c / 3, v = c % 3;
    const int yy = py + u - 1, xx = px + v - 1;
    mp[c] = ((unsigned)yy < 64u && (unsigned)xx < 64u) ? Mk[yy * 64 + xx] : 0.f;
  }
  __syncthreads();

  float q[9], k[9], v[9];
  const float bqc = bq[c], bkc = bk[c], bvc = bv[c];
#pragma unroll
  for (int p = 0; p < 9; ++p) { q[p] = bqc; k[p] = bkc; v[p] = bvc; }
  for (int c2 = 0; c2 < 64; ++c2) {
    const float wq = WqT[c2 * 64 + c], wk = WkT[c2 * 64 + c], wv = WvT[c2 * 64 + c];
#pragma unroll
    for (int p = 0; p < 9; ++p) {
      const float xv = ckp[c2][p];
      q[p] += wq * xv; k[p] += wk * xv; v[p] += wv * xv;
    }
  }
#pragma unroll
  for (int p = 0; p < 9; ++p) ks[c][p] = k[p];
  __syncthreads();
  if (c < 9) { float s = 0.f; for (int d = 0; d < 64; ++d) s += ks[d][c]; ksum[c] = s; }
  __syncthreads();

  float sdot = 0.f;
#pragma unroll
  for (int p = 0; p < 9; ++p) sdot += q[p] * ksum[p];

  float xs[9], mx = -3.4e38f;
#pragma unroll
  for (int p = 0; p < 9; ++p) { xs[p] = mp[p] * sdot; mx = fmaxf(mx, xs[p]); }
  float tot = 0.f;
#pragma unroll
  for (int p = 0; p < 9; ++p) { xs[p] = __expf(xs[p] - mx); tot += xs[p]; }
  const float einv = 1.f / tot;
  const float g = gamma[0];

  float cko[9], ssq = 0.f;
#pragma unroll
  for (int p = 0; p < 9; ++p) {
    const float att = xs[p] * einv * v[p];
    const float mpp = mp[p];
    const float o = ckp[c][p] * mpp + g * (1.f - mpp) * att;
    cko[p] = o; ssq += o * o;
  }
  red[c] = ssq; __syncthreads();
  for (int s = 32; s > 0; s >>= 1) { if (c < s) red[c] += red[c + s]; __syncthreads(); }
  const float rinv = rsqrtf(red[0]);

  float* o = ckmat + (size_t)n * 576 + c * 9;
#pragma unroll
  for (int p = 0; p < 9; ++p) o[p] = cko[p] * rinv;
  if (c == 0) {
    float msum = 0.f;
#pragma unroll
    for (int p = 0; p < 9; ++p) msum += mp[p];
    mmv[n] = (msum == 0.f) ? 1.f : 0.f;
  }
}

// rec[c][y][x] = sum_{u,v} S[(y+u-1,x+v-1)][c*9 + (2-u)*3 + (2-v)]  (conv_transpose)
__global__ void rec_gather(const float* __restrict__ S, float* __restrict__ out)
{
  const int idx = blockIdx.x * 256 + threadIdx.x;  // 64*4096
  const int c = idx >> 12, pos = idx & 4095;
  const int y = pos >> 6, x = pos & 63;
  float acc = 0.f;
#pragma unroll
  for (int u = 0; u < 3; ++u)
#pragma unroll
    for (int v = 0; v < 3; ++v) {
      const int yy = y + u - 1, xx = x + v - 1;
      if ((unsigned)yy < 64u && (unsigned)xx < 64u)
        acc += S[(size_t)(yy * 64 + xx) * 576 + c * 9 + (2 - u) * 3 + (2 - v)];
    }
  out[idx] = acc;
}

// ---------------------------------------------------------------------------
extern "C" void kernel_launch(void* const* d_in, const int* in_sizes, int n_in,
                              void* d_out, int out_size, void* d_ws, size_t ws_size,
                              hipStream_t stream)
{
  (void)in_sizes; (void)n_in; (void)out_size; (void)ws_size;
  const float* fgnd = (const float*)d_in[0];
  const float* mask = (const float*)d_in[1];
  const float* Wq_f = (const float*)d_in[3];
  const float* bq_f = (const float*)d_in[4];
  const float* Wk_f = (const float*)d_in[5];
  const float* bk_f = (const float*)d_in[6];
  const float* Wv_f = (const float*)d_in[7];
  const float* bv_f = (const float*)d_in[8];
  const float* Wq_p = (const float*)d_in[9];
  const float* bq_p = (const float*)d_in[10];
  const float* Wk_p = (const float*)d_in[11];
  const float* bk_p = (const float*)d_in[12];
  const float* Wv_p = (const float*)d_in[13];
  const float* bv_p = (const float*)d_in[14];
  const float* gam  = (const float*)d_in[15];
  float* out = (float*)d_out;
  float* ws  = (float*)d_ws;

  // workspace offsets (floats); total 25,444,352 floats ≈ 102 MB
  float* bg     = ws + 0;
  float* qp     = ws + 262144;
  float* kp     = ws + 524288;
  float* vp     = ws + 786432;
  float* attout = ws + 1048576;
  float* fgb    = ws + 1310720;
  float* ckmat  = ws + 1572864;   // 4096*576
  float* fmcol  = ws + 3932160;   // 4096*576
  float* Sbuf   = ws + 6291456;   // 4096*576
  float* mm     = ws + 8650752;   // 4096
  float* wT     = ws + 8654848;   // 3*4096
  float* big    = ws + 8667136;   // 4096*4096

  transpose_w<<<48, 256, 0, stream>>>(Wq_p, Wk_p, Wv_p, wT);

  for (int b = 0; b < 2; ++b) {
    const float* F  = fgnd + (size_t)b * 64 * 4096;
    const float* Mk = mask + (size_t)b * 4096;

    // ---- MPGA global attention -------------------------------------------
    make_background<<<1024, 256, 0, stream>>>(F, Mk, bg);
    gemm_wmma_f32<<<dim3(64, 1), 128, 0, stream>>>(Wq_f, F,  bq_f, qp, 64, 4096, 64, 0, 0);
    gemm_wmma_f32<<<dim3(64, 1), 128, 0, stream>>>(Wk_f, bg, bk_f, kp, 64, 4096, 64, 0, 0);
    gemm_wmma_f32<<<dim3(64, 1), 128, 0, stream>>>(Wv_f, bg, bv_f, vp, 64, 4096, 64, 0, 0);
    // scores[i][j] = sum_c qp[c][i]*kp[c][j]
    gemm_wmma_f32<<<dim3(64, 64), 128, 0, stream>>>(qp, kp, nullptr, big, 4096, 4096, 64, 1, 0);
    row_softmax_mask<<<4096, 256, 0, stream>>>(big, Mk, 4096);
    // out[c][i] = sum_j vp[c][j]*A[i][j]
    gemm_wmma_f32<<<dim3(64, 1), 128, 0, stream>>>(vp, big, nullptr, attout, 64, 4096, 4096, 0, 1);
    compose_fg<<<1024, 256, 0, stream>>>(F, Mk, attout, fgb);

    // ---- patch GAP + normalization ---------------------------------------
    gap_patches<<<4096, 64, 0, stream>>>(bg, Mk, wT, wT + 4096, wT + 8192,
                                         bq_p, bk_p, bv_p, gam, ckmat, mm);

    // ---- contextual attention convs --------------------------------------
    im2col3<<<1024, 256, 0, stream>>>(fgb, fmcol);
    // cr[n][p] = <ckmat[n], fmcol[p]>
    gemm_wmma_f32<<<dim3(64, 64), 128, 0, stream>>>(ckmat, fmcol, nullptr, big, 4096, 4096, 576, 0, 1);
    box3x3_inplace<<<4096, 256, 0, stream>>>(big);
    col_softmax_mm<<<16, 256, 0, stream>>>(big, mm);
    // S[p][cuv] = sum_n att[n][p]*ckmat[n][cuv]
    gemm_wmma_f32<<<dim3(9, 64), 128, 0, stream>>>(big, ckmat, nullptr, Sbuf, 4096, 576, 4096, 1, 0);
    rec_gather<<<1024, 256, 0, stream>>>(Sbuf, out + (size_t)b * 64 * 4096);
  }
}